// Compute_fg_bg_similarity_44281112822069
// MI455X (gfx1250) — compile-verified
//
#include <hip/hip_runtime.h>
#include <cstdint>
#include <cstddef>

// Problem constants (match reference)
#define IMG_H 256
#define IMG_W 256
#define NB    16
#define HW    (IMG_H * IMG_W)

// Tiling
#define TS     32                // tile side
#define HALO   (TS + 2)          // 34
#define PLANE  (HALO * HALO)     // 1156
#define NPLANES 9                // 3 lab + 6 vec channels
#define NTHREADS 256

// ---------------------------------------------------------------------------
// CDNA5 async global->LDS copy (ASYNCcnt path). Builtin if the toolchain has
// it, otherwise literal gfx1250 inline asm per CDNA5 ISA.
// Builtin signature (per hipcc diagnostic): pointee type is `int`.
// ---------------------------------------------------------------------------
__device__ __forceinline__ void async_load_f32(const float* g, float* l) {
#if __has_builtin(__builtin_amdgcn_global_load_async_to_lds_b32)
  __builtin_amdgcn_global_load_async_to_lds_b32(
      (__attribute__((address_space(1))) int*)const_cast<float*>(g),
      (__attribute__((address_space(3))) int*)l,
      /*offset=*/0, /*cpol=*/0);
#else
  unsigned l32 = (unsigned)(unsigned long long)
      (__attribute__((address_space(3))) char*)l;
  asm volatile("global_load_async_to_lds_b32 %0, %1, off"
               :
               : "v"(l32), "v"(g)
               : "memory");
#endif
}

__device__ __forceinline__ void async_wait_all() {
#if __has_builtin(__builtin_amdgcn_s_wait_asynccnt)
  __builtin_amdgcn_s_wait_asynccnt(0);
#else
  asm volatile("s_wait_asynccnt 0" ::: "memory");
#endif
}

// ---------------------------------------------------------------------------
// Kernel 1: per-tile stencil. One 32x32 tile per block, 256 threads (8 waves).
// ---------------------------------------------------------------------------
__global__ __launch_bounds__(NTHREADS)
void sim_maps_kernel(const float* __restrict__ lab,    // (B,3,H,W)
                     const float* __restrict__ vec,    // (B,6,H,W)
                     float* __restrict__ out,          // (B,16,H,W)
                     float* __restrict__ partials) {   // per-block loss partial
  __shared__ float sm[NPLANES * PLANE];  // 9 halo planes: 41,616 B
  __shared__ float red[NTHREADS];

  const int tid = threadIdx.x;
  const int tx0 = blockIdx.x * TS;
  const int ty0 = blockIdx.y * TS;
  const int b   = blockIdx.z;

  const float* labB = lab + (size_t)b * 3 * HW;
  const float* vecB = vec + (size_t)b * 6 * HW;

  // ---- Stage 9 halo planes into LDS via async DMA; zero-fill OOB taps ----
  for (int i = tid; i < NPLANES * PLANE; i += NTHREADS) {
    const int p   = i / PLANE;
    const int rem = i - p * PLANE;
    const int r   = rem / HALO;
    const int c   = rem - r * HALO;
    const int gy  = ty0 + r - 1;
    const int gx  = tx0 + c - 1;
    if ((unsigned)gy < (unsigned)IMG_H && (unsigned)gx < (unsigned)IMG_W) {
      const float* src = (p < 3)
          ? (labB + (size_t)p * HW + (size_t)gy * IMG_W + gx)
          : (vecB + (size_t)(p - 3) * HW + (size_t)gy * IMG_W + gx);
      async_load_f32(src, &sm[i]);
    } else {
      sm[i] = 0.0f;  // reference zero-pads the border
    }
  }
  async_wait_all();   // drain this wave's ASYNCcnt
  __syncthreads();    // all waves' tiles visible in LDS

  // ---- Compute: each thread owns 4 pixels (x = tid&31, y = tid>>5 + 8k) ----
  const int tx  = tid & (TS - 1);
  const int tys = tid >> 5;

  // neighbor order matches the reference's unfold_wo_center enumeration
  const int dy[8] = {-1, -1, -1,  0, 0,  1, 1, 1};
  const int dx[8] = {-1,  0,  1, -1, 1, -1, 0, 1};

  float acc = 0.0f;

#pragma unroll
  for (int k = 0; k < 4; ++k) {
    const int ty = tys + (k << 3);
    const int r  = ty + 1;
    const int c  = tx + 1;
    const int ctr = r * HALO + c;

    const float L0 = sm[0 * PLANE + ctr];
    const float L1 = sm[1 * PLANE + ctr];
    const float L2 = sm[2 * PLANE + ctr];
    const float V0 = sm[3 * PLANE + ctr];
    const float V1 = sm[4 * PLANE + ctr];
    const float V2 = sm[5 * PLANE + ctr];
    const float V3 = sm[6 * PLANE + ctr];
    const float V4 = sm[7 * PLANE + ctr];
    const float V5 = sm[8 * PLANE + ctr];

    const size_t obase =
        (size_t)b * NB * HW + (size_t)(ty0 + ty) * IMG_W + (size_t)(tx0 + tx);

#pragma unroll
    for (int n = 0; n < 8; ++n) {
      const int o = (r + dy[n]) * HALO + (c + dx[n]);

      const float d0 = L0 - sm[0 * PLANE + o];
      const float d1 = L1 - sm[1 * PLANE + o];
      const float d2 = L2 - sm[2 * PLANE + o];
      const float dl = d0 * d0 + d1 * d1 + d2 * d2;
      const float w  = expf(-sqrtf(dl));          // alpha_3 = 1.0

      const float e0 = V0 - sm[3 * PLANE + o];
      const float e1 = V1 - sm[4 * PLANE + o];
      const float e2 = V2 - sm[5 * PLANE + o];
      const float e3 = V3 - sm[6 * PLANE + o];
      const float e4 = V4 - sm[7 * PLANE + o];
      const float e5 = V5 - sm[8 * PLANE + o];
      const float dv = e0 * e0 + e1 * e1 + e2 * e2 +
                       e3 * e3 + e4 * e4 + e5 * e5;
      const float v  = sqrtf(dv);

      // streaming (write-once) output: non-temporal stores
      __builtin_nontemporal_store(w, out + obase + (size_t)n * HW);
      __builtin_nontemporal_store(v, out + obase + (size_t)(8 + n) * HW);

      if (w >= 0.1f) acc += v;                    // vetorch_thresh = 0.1
    }
  }

  // ---- Deterministic fixed-order block reduction of the loss partial ----
  red[tid] = acc;
  __syncthreads();
#pragma unroll
  for (int off = NTHREADS / 2; off > 0; off >>= 1) {
    if (tid < off) red[tid] += red[tid + off];
    __syncthreads();
  }
  if (tid == 0) {
    const int bid = (blockIdx.z * gridDim.y + blockIdx.y) * gridDim.x + blockIdx.x;
    partials[bid] = red[0];
  }
}

// ---------------------------------------------------------------------------
// Kernel 2: deterministic final reduction (fixed order, double accumulation).
// ---------------------------------------------------------------------------
__global__ __launch_bounds__(NTHREADS)
void loss_reduce_kernel(const float* __restrict__ partials, int n,
                        float* __restrict__ loss_out) {
  __shared__ double red[NTHREADS];
  double s = 0.0;
  for (int i = threadIdx.x; i < n; i += NTHREADS) s += (double)partials[i];
  red[threadIdx.x] = s;
  __syncthreads();
#pragma unroll
  for (int off = NTHREADS / 2; off > 0; off >>= 1) {
    if (threadIdx.x < off) red[threadIdx.x] += red[threadIdx.x + off];
    __syncthreads();
  }
  if (threadIdx.x == 0) {
    // mean over B*8*H*W = 16*8*256*256 = 8,388,608 elements
    loss_out[0] = (float)(red[0] / 8388608.0);
  }
}

// ---------------------------------------------------------------------------
// Launch: d_in[0] = images_lab (16,3,256,256) f32
//         d_in[1] = pixel_vector (16,6,256,256) f32
//         d_out   = sim_maps (16,16,256,256) f32  ++  neighbor_loss (1) f32
//         d_ws    = 1024 per-block partials (4 KB)
// ---------------------------------------------------------------------------
extern "C" void kernel_launch(void* const* d_in, const int* in_sizes, int n_in,
                              void* d_out, int out_size, void* d_ws, size_t ws_size,
                              hipStream_t stream) {
  (void)in_sizes; (void)n_in; (void)out_size; (void)ws_size;

  const float* lab = (const float*)d_in[0];
  const float* vec = (const float*)d_in[1];
  float* out       = (float*)d_out;
  float* partials  = (float*)d_ws;   // 1024 floats

  dim3 grid(IMG_W / TS, IMG_H / TS, 16);   // (8, 8, 16) = 1024 blocks
  sim_maps_kernel<<<grid, NTHREADS, 0, stream>>>(lab, vec, out, partials);

  const int nblocks = (IMG_W / TS) * (IMG_H / TS) * 16;  // 1024
  loss_reduce_kernel<<<1, NTHREADS, 0, stream>>>(
      partials, nblocks, out + (size_t)16 * NB * HW);    // d_out[16777216]
}